// ZBLRepulsionEnergy_26534307955286
// MI455X (gfx1250) — compile-verified
//
#include <hip/hip_runtime.h>
#include <hip/hip_bf16.h>
#include <math.h>

#define RCUT     5.2f
#define KE_HALF  7.199822675975274f
#define LOG2E    1.4426950408889634f

// Try the CDNA5 async global->LDS DMA for table staging; flip to 0 if the
// assembler rejects the mnemonic.
#ifndef ZBL_ASYNC_STAGE
#define ZBL_ASYNC_STAGE 1
#endif

// ---------------------------------------------------------------------------
// Kernel 1: one-time parameter prep (1 block, runs in ~us; SALU-heavy is fine).
// ws layout (floats): [0..3]=c/sum(c)  [4..7]=-d_k*log2(e)  [8]=zbl_pow
//                     [9]=1/zbl_length [10..128]= z^zbl_pow table (z=0..118)
// ---------------------------------------------------------------------------
__global__ void zbl_params_kernel(const float* __restrict__ c_raw,
                                  const float* __restrict__ d_raw,
                                  const float* __restrict__ pow_raw,
                                  const float* __restrict__ len_raw,
                                  float* __restrict__ params) {
    const int t = threadIdx.x;
    float c[4], d[4], csum = 0.0f;
    #pragma unroll
    for (int k = 0; k < 4; ++k) { c[k] = log1pf(expf(c_raw[k])); csum += c[k]; }
    #pragma unroll
    for (int k = 0; k < 4; ++k) { d[k] = log1pf(expf(d_raw[k])); }
    const float zpow = log1pf(expf(pow_raw[0]));
    const float zlen = log1pf(expf(len_raw[0]));

    if (t < 4) {
        params[t]     = c[t] / csum;
        params[4 + t] = -d[t] * LOG2E;   // exp(-d*x) == exp2f(param * x)
    }
    if (t == 4) params[8] = zpow;
    if (t == 5) params[9] = 1.0f / zlen;
    if (t < 119) params[10 + t] = powf((float)t, zpow);
}

__global__ void zbl_zero_kernel(float* __restrict__ out, int n) {
    int i = blockIdx.x * blockDim.x + threadIdx.x;
    if (i < n) out[i] = 0.0f;
}

// ---------------------------------------------------------------------------
// LDS staging: s_tab[z] = {covalent_radii[z], z^zbl_pow}
// Uses CDNA5 async global->LDS DMA (ASYNCcnt) when enabled.
// ---------------------------------------------------------------------------
__device__ __forceinline__ void zbl_stage_tables(float2* s_tab,
                                                 const float* __restrict__ covr,
                                                 const float* __restrict__ params) {
    for (int t = threadIdx.x; t < 119; t += blockDim.x) {
#if ZBL_ASYNC_STAGE
        unsigned lds_lo = (unsigned)(uintptr_t)(&s_tab[t].x);
        unsigned lds_hi = (unsigned)(uintptr_t)(&s_tab[t].y);
        asm volatile("global_load_async_to_lds_b32 %0, %1, off"
                     :: "v"(lds_lo), "v"(covr + t) : "memory");
        asm volatile("global_load_async_to_lds_b32 %0, %1, off"
                     :: "v"(lds_hi), "v"(params + 10 + t) : "memory");
#else
        s_tab[t] = make_float2(covr[t], params[10 + t]);
#endif
    }
#if ZBL_ASYNC_STAGE
    asm volatile("s_wait_asynccnt 0" ::: "memory");
#endif
    __syncthreads();
}

// one edge's full math + conditional atomic scatter
__device__ __forceinline__ void zbl_edge_body(float r, int i, int j,
                                              const int* __restrict__ anum,
                                              const float2* s_tab,
                                              float c0, float c1, float c2, float c3,
                                              float nd0, float nd1, float nd2, float nd3,
                                              float inv_zl,
                                              float* __restrict__ zbl_out) {
    const int ni = anum[i];
    const int nj = anum[j];
    const float2 ti = s_tab[ni];
    const float2 tj = s_tab[nj];

    const float ra = r * (ti.y + tj.y) * inv_zl;            // lengths / a
    const float f  = fmaf(c0, exp2f(nd0 * ra),
                     fmaf(c1, exp2f(nd1 * ra),
                     fmaf(c2, exp2f(nd2 * ra), c3 * exp2f(nd3 * ra))));

    const float rc = ti.x + tj.x;
    const float x  = r * __builtin_amdgcn_rcpf(rc);
    if ((r <= RCUT) && (x < 1.0f)) {
        const float x2  = x * x;
        const float x6  = x2 * x2 * x2;
        // 1 - 28 x^6 + 48 x^7 - 21 x^8
        const float env = fmaf(x6, fmaf(x, fmaf(-21.0f, x, 48.0f), -28.0f), 1.0f);
        const float pref = KE_HALF * (float)(ni * nj) * __builtin_amdgcn_rcpf(r);
        unsafeAtomicAdd(&zbl_out[i], pref * f * env);       // global_atomic_add_f32
    }
}

// ---------------------------------------------------------------------------
// Kernel 3a: vectorized main loop, 4 edges / thread-iteration (b128 loads).
// Requires n_edges even (16B alignment of the j-row) — guaranteed by caller.
// ---------------------------------------------------------------------------
__global__ __launch_bounds__(256) void zbl_edge_kernel_v4(
        const float4* __restrict__ len4,
        const uint4*  __restrict__ ei4,   // i-row int64s as u32 pairs (2 edges/uint4)
        const uint4*  __restrict__ ej4,   // j-row, pre-offset by caller
        const int*    __restrict__ anum,
        const float*  __restrict__ covr,
        const float*  __restrict__ params,
        float*        __restrict__ zbl_out,
        int n_quads) {
    __shared__ float2 s_tab[128];
    zbl_stage_tables(s_tab, covr, params);

    const float c0 = params[0], c1 = params[1], c2 = params[2], c3 = params[3];
    const float nd0 = params[4], nd1 = params[5], nd2 = params[6], nd3 = params[7];
    const float inv_zl = params[9];

    const int stride = gridDim.x * blockDim.x;
    for (int q = blockIdx.x * blockDim.x + threadIdx.x; q < n_quads; q += stride) {
        const int qp = q + stride;
        if (qp < n_quads) {
            __builtin_prefetch(&len4[qp], 0, 1);            // global_prefetch_b8
            __builtin_prefetch(&ei4[2 * (size_t)qp], 0, 1);
            __builtin_prefetch(&ej4[2 * (size_t)qp], 0, 1);
        }
        const float4 rv = len4[q];
        const uint4  ia = ei4[2 * (size_t)q];
        const uint4  ib = ei4[2 * (size_t)q + 1];
        const uint4  ja = ej4[2 * (size_t)q];
        const uint4  jb = ej4[2 * (size_t)q + 1];

        const float rr[4]  = { rv.x, rv.y, rv.z, rv.w };
        const int   ii[4]  = { (int)ia.x, (int)ia.z, (int)ib.x, (int)ib.z };
        const int   jj[4]  = { (int)ja.x, (int)ja.z, (int)jb.x, (int)jb.z };
        #pragma unroll
        for (int k = 0; k < 4; ++k) {
            zbl_edge_body(rr[k], ii[k], jj[k], anum, s_tab,
                          c0, c1, c2, c3, nd0, nd1, nd2, nd3, inv_zl, zbl_out);
        }
    }
}

// ---------------------------------------------------------------------------
// Kernel 3b: scalar path (tail edges, or everything when n_edges is odd).
// ---------------------------------------------------------------------------
__global__ __launch_bounds__(256) void zbl_edge_kernel_scalar(
        const float*    __restrict__ lengths,
        const unsigned* __restrict__ eidx32,
        const int*      __restrict__ anum,
        const float*    __restrict__ covr,
        const float*    __restrict__ params,
        float*          __restrict__ zbl_out,
        int start, int n_edges) {
    __shared__ float2 s_tab[128];
    zbl_stage_tables(s_tab, covr, params);

    const float c0 = params[0], c1 = params[1], c2 = params[2], c3 = params[3];
    const float nd0 = params[4], nd1 = params[5], nd2 = params[6], nd3 = params[7];
    const float inv_zl = params[9];

    const int stride = gridDim.x * blockDim.x;
    for (int e = start + blockIdx.x * blockDim.x + threadIdx.x; e < n_edges; e += stride) {
        const float r = lengths[e];
        const int   i = (int)eidx32[2 * (size_t)e];
        const int   j = (int)eidx32[2 * ((size_t)n_edges + e)];
        zbl_edge_body(r, i, j, anum, s_tab,
                      c0, c1, c2, c3, nd0, nd1, nd2, nd3, inv_zl, zbl_out);
    }
}

__global__ void zbl_final_kernel(const float* __restrict__ ae,
                                 float* __restrict__ out, int n) {
    int i = blockIdx.x * blockDim.x + threadIdx.x;
    if (i < n) out[n + i] = ae[i] + out[i];
}

extern "C" void kernel_launch(void* const* d_in, const int* in_sizes, int n_in,
                              void* d_out, int out_size, void* d_ws, size_t ws_size,
                              hipStream_t stream) {
    const float*     lengths  = (const float*)d_in[0];
    const float*     ae       = (const float*)d_in[1];
    const float*     c_raw    = (const float*)d_in[2];
    const float*     d_raw    = (const float*)d_in[3];
    const float*     pow_raw  = (const float*)d_in[4];
    const float*     len_raw  = (const float*)d_in[5];
    const float*     covr     = (const float*)d_in[6];
    const int*       anum     = (const int*)d_in[7];
    const unsigned*  eidx32   = (const unsigned*)d_in[8];   // int64 pairs; low dwords used

    const int n_edges = in_sizes[0];
    const int n_atoms = in_sizes[1];
    float* out    = (float*)d_out;
    float* params = (float*)d_ws;   // 129 floats used

    zbl_params_kernel<<<1, 128, 0, stream>>>(c_raw, d_raw, pow_raw, len_raw, params);
    zbl_zero_kernel<<<(n_atoms + 255) / 256, 256, 0, stream>>>(out, n_atoms);

    if (n_edges >= 8 && (n_edges & 1) == 0) {
        const int n_quads = n_edges >> 2;          // vector part covers 4*n_quads edges
        int blocks = (n_quads + 255) / 256;
        if (blocks > 4096) blocks = 4096;
        zbl_edge_kernel_v4<<<blocks, 256, 0, stream>>>(
            (const float4*)lengths,
            (const uint4*)eidx32,
            (const uint4*)(eidx32 + 2 * (size_t)n_edges),
            anum, covr, params, out, n_quads);
        const int base = n_quads << 2;
        if (base < n_edges) {                      // <4 leftover edges
            zbl_edge_kernel_scalar<<<1, 256, 0, stream>>>(
                lengths, eidx32, anum, covr, params, out, base, n_edges);
        }
    } else if (n_edges > 0) {
        int blocks = (n_edges + 255) / 256;
        if (blocks > 4096) blocks = 4096;
        zbl_edge_kernel_scalar<<<blocks, 256, 0, stream>>>(
            lengths, eidx32, anum, covr, params, out, 0, n_edges);
    }

    zbl_final_kernel<<<(n_atoms + 255) / 256, 256, 0, stream>>>(ae, out, n_atoms);
}